// VQVAE_17428977287173
// MI455X (gfx1250) — compile-verified
//
#include <hip/hip_runtime.h>
#include <hip/hip_bf16.h>
#include <math.h>

typedef __attribute__((ext_vector_type(2))) float v2f;
typedef __attribute__((ext_vector_type(8))) float v8f;

#define N_PTS 65536
#define DIM 64
#define KCB 1024
#define INV_TEMP 100.0f   // 1/ENTROPY_TEMPERATURE
#define ROWS_PB 32        // rows per block (2 row-tiles, 128 KB LDS on CDNA5)

// workspace layout (float offsets)
#define WS_B2     0       // [0,1024)    codebook squared norms
#define WS_PROBS  1024    // [1024,2048) avg-prob accumulators
#define WS_SUMMIN 2048    // scalar: sum of per-row min distances
#define WS_SUMH   2049    // scalar: sum of per-row entropies
#define WS_IDX    2112    // [2112, 2112+65536) as int: argmin indices

// ---------------------------------------------------------------------------
// Kernel 0: b2[k] = ||codebook[k]||^2, and zero the accumulators (every call).
// ---------------------------------------------------------------------------
__global__ void vq_b2_init(const float* __restrict__ cb, float* __restrict__ ws) {
    int k = blockIdx.x * blockDim.x + threadIdx.x;
    if (k < KCB) {
        const float* r = cb + k * DIM;
        float s = 0.f;
        #pragma unroll
        for (int j = 0; j < DIM; ++j) s += r[j] * r[j];
        ws[WS_B2 + k] = s;
        ws[WS_PROBS + k] = 0.f;
        if (k == 0) { ws[WS_SUMMIN] = 0.f; ws[WS_SUMH] = 0.f; }
    }
}

// ---------------------------------------------------------------------------
// Kernel 1: 32-row tile per block. fp32 WMMA GEMM vs whole codebook with
// B-fragment register reuse across the two 16-row tiles (codebook read once
// per block -> halved L2 traffic vs 16-row blocks). Distances staged in
// 128 KB LDS (CDNA5 320 KB/WGP), then argmin + softmax stats + avg-probs.
// Block = 256 threads = 8 wave32's; wave w covers columns [w*128, w*128+128).
// a2 values for the rows this lane writes are hoisted to registers so the
// WMMA epilogue has no LDS loads / dscnt waits in the hot loop.
// ---------------------------------------------------------------------------
__global__ __launch_bounds__(256) void vq_main(const float* __restrict__ x,
                                               const float* __restrict__ cb,
                                               float* __restrict__ ws) {
    __shared__ float dist[ROWS_PB][KCB];   // 128 KB
    __shared__ float a2s[ROWS_PB];
    __shared__ float rzrow[ROWS_PB];

    const int tid  = threadIdx.x;
    const int lane = tid & 31;
    const int wave = tid >> 5;
    const int lm   = lane & 15;   // row (A) / col (B) within 16-tile
    const int h    = lane >> 4;   // lane half selects K pair {2h, 2h+1}
    const int rowbase = blockIdx.x * ROWS_PB;

    // ---- A fragments for both 16x64 row tiles (reused across all col tiles)
    // 16x4 f32 A layout: lanes 0-15 row M=lane, K=0,1; lanes 16-31 row M, K=2,3
    v2f a[2][16];
    #pragma unroll
    for (int rt = 0; rt < 2; ++rt) {
        const float* xr = x + (size_t)(rowbase + rt * 16 + lm) * DIM + h * 2;
        float a2p = 0.f;
        #pragma unroll
        for (int kk = 0; kk < 16; ++kk) {
            v2f t; t.x = xr[kk * 4]; t.y = xr[kk * 4 + 1];
            a[rt][kk] = t;
            a2p += t.x * t.x + t.y * t.y;
        }
        a2p += __shfl_xor(a2p, 16);              // combine both K halves
        if (tid < 16) a2s[rt * 16 + lm] = a2p;   // ||x_row||^2
    }
    __syncthreads();

    // Hoist the 16 a2 values this lane's D-fragment rows need into registers
    // (invariant across all 8 column tiles -> no LDS traffic in the hot loop).
    float a2r[2][8];
    #pragma unroll
    for (int rt = 0; rt < 2; ++rt)
        #pragma unroll
        for (int v = 0; v < 8; ++v)
            a2r[rt][v] = a2s[rt * 16 + h * 8 + v];

    const float* b2g = ws + WS_B2;
    for (int t = 0; t < 8; ++t) {
        const int colbase = (wave * 8 + t) * 16;
        const float* cbr = cb + (size_t)(colbase + lm) * DIM + h * 2;
        // B fragments for this col tile, loaded ONCE, reused for both row tiles
        v2f b[16];
        #pragma unroll
        for (int kk = 0; kk < 16; ++kk) {
            v2f tt; tt.x = cbr[kk * 4]; tt.y = cbr[kk * 4 + 1];
            b[kk] = tt;
        }
        const float b2n = b2g[colbase + lm];
        #pragma unroll
        for (int rt = 0; rt < 2; ++rt) {
            v8f c = {};
            #pragma unroll
            for (int kk = 0; kk < 16; ++kk) {
                // D = A(16x4,f32) x B(4x16,f32) + C -> v_wmma_f32_16x16x4_f32
                c = __builtin_amdgcn_wmma_f32_16x16x4_f32(
                        false, a[rt][kk], false, b[kk], (short)0, c, false, false);
            }
            const int m0 = rt * 16 + h * 8;      // D layout: M = v + 8*half
            #pragma unroll
            for (int v = 0; v < 8; ++v)
                dist[m0 + v][colbase + lm] = a2r[rt][v] - 2.f * c[v] + b2n;
        }
    }
    __syncthreads();

    // ---- per-row min/argmin + softmax stats: 16 threads per row ------------
    const int lg = tid & 15;
    #pragma unroll
    for (int ro = 0; ro < 2; ++ro) {
        const int row = ro * 16 + (tid >> 4);
        float mn = 3.4e38f; int mi = 0;
        for (int j = 0; j < 64; ++j) {
            const int k = lg + 16 * j;
            const float d = dist[row][k];
            if (d < mn) { mn = d; mi = k; }
        }
        #pragma unroll
        for (int m = 8; m >= 1; m >>= 1) {       // butterfly in 16-lane group
            float ov = __shfl_xor(mn, m);
            int   oi = __shfl_xor(mi, m);
            if (ov < mn || (ov == mn && oi < mi)) { mn = ov; mi = oi; }
        }
        if (lg == 0) {
            ((int*)ws)[WS_IDX + rowbase + row] = mi;
            atomicAdd(ws + WS_SUMMIN, mn);       // min dist == ||q - x||^2
        }
        // softmax stats: logits l = -100*d, row max = -100*mn
        float Z = 0.f, S1 = 0.f;
        for (int j = 0; j < 64; ++j) {
            const int k = lg + 16 * j;
            const float tt = -INV_TEMP * (dist[row][k] - mn);   // <= 0
            const float e  = __expf(tt);
            Z += e; S1 += tt * e;
            dist[row][k] = e;                    // stash unnormalized prob
        }
        #pragma unroll
        for (int m = 8; m >= 1; m >>= 1) { Z += __shfl_xor(Z, m); S1 += __shfl_xor(S1, m); }
        if (lg == 0) {
            rzrow[row] = 1.f / Z;
            atomicAdd(ws + WS_SUMH, logf(Z) - S1 / Z);  // row entropy
        }
    }
    __syncthreads();

    // ---- avg-prob accumulation: one partial per (block, k), then atomic ----
    #pragma unroll
    for (int q = 0; q < 4; ++q) {
        const int k = tid + q * 256;
        float s = 0.f;
        #pragma unroll
        for (int r = 0; r < ROWS_PB; ++r) s += dist[r][k] * rzrow[r];
        atomicAdd(ws + WS_PROBS + k, s);
    }
}

// ---------------------------------------------------------------------------
// Kernel 2: gather quantized rows (straight-through fwd == gather) + indices.
// ---------------------------------------------------------------------------
__global__ void vq_gather(const float* __restrict__ cb, const int* __restrict__ idx,
                          float* __restrict__ out) {
    const int i = blockIdx.x * blockDim.x + threadIdx.x;   // i < N*D
    const int r = i >> 6, c = i & 63;
    out[i] = cb[(size_t)idx[r] * DIM + c];
    if (i < N_PTS) out[(size_t)N_PTS * DIM + 1 + i] = (float)idx[i];
}

// ---------------------------------------------------------------------------
// Kernel 3: final scalar loss.
// ---------------------------------------------------------------------------
__global__ void vq_loss(const float* __restrict__ ws, float* __restrict__ out) {
    __shared__ float red[256];
    const int tid = threadIdx.x;
    float loc = 0.f;
    for (int k = tid; k < KCB; k += 256) {
        const float a = ws[WS_PROBS + k] * (1.0f / (float)N_PTS);
        loc += a * logf(a + 1e-5f);
    }
    red[tid] = loc;
    __syncthreads();
    for (int s = 128; s >= 1; s >>= 1) {
        if (tid < s) red[tid] += red[tid + s];
        __syncthreads();
    }
    if (tid == 0) {
        const float avg_entropy    = -red[0];
        const float sample_entropy = ws[WS_SUMH] / (float)N_PTS;
        const float mse            = ws[WS_SUMMIN] / ((float)N_PTS * (float)DIM);
        // e_latent(0.25*mse) + q_latent(mse) + 0.1*(sample_H - avg_H)
        out[(size_t)N_PTS * DIM] = 1.25f * mse + 0.1f * (sample_entropy - avg_entropy);
    }
}

extern "C" void kernel_launch(void* const* d_in, const int* in_sizes, int n_in,
                              void* d_out, int out_size, void* d_ws, size_t ws_size,
                              hipStream_t stream) {
    (void)in_sizes; (void)n_in; (void)out_size; (void)ws_size;
    const float* x  = (const float*)d_in[0];   // (64,32,32,64) fp32
    const float* cb = (const float*)d_in[1];   // (1024,64)    fp32
    float* ws  = (float*)d_ws;
    float* out = (float*)d_out;

    vq_b2_init<<<4, 256, 0, stream>>>(cb, ws);
    vq_main  <<<N_PTS / ROWS_PB, 256, 0, stream>>>(x, cb, ws);
    vq_gather<<<(N_PTS * DIM) / 256, 256, 0, stream>>>(cb, (const int*)ws + WS_IDX, out);
    vq_loss  <<<1, 256, 0, stream>>>(ws, out);
}